// DeBut_1dConv_65515431133410
// MI455X (gfx1250) — compile-verified
//
#include <hip/hip_runtime.h>

// DeBut 1D-conv chain for MI455X (gfx1250, wave32, WMMA).
//
//   x:(16,512,2048) f32, twiddle:(40960,) f32, bias:(512,) f32
//   rows = im2col(x) : (16*2045, 2048), feature j = k*512 + c -> x[b,c,l+k]
//   chain widths: 2048 -> 2048 -> 2048 -> 1024 -> 512
//   out:(16,512,2045) f32
//
// One 16-row tile per workgroup (256 thr = 8 waves, wave32). Tile lives in
// LDS; each butterfly factor = batched (M=16, N=row, K=col) GEMMs on
// V_WMMA_F32_16X16X4_F32. Per-factor twiddle slabs are bulk-staged into LDS
// (coalesced float4) so the WMMA dependence chain only sees LDS latency.

typedef float v2f __attribute__((ext_vector_type(2)));
typedef float v8f __attribute__((ext_vector_type(8)));

#define L_OUT   2045
#define N_TILES 2045
#define SA      2052            // bufA row stride (floats): bank-spread 16 rows
#define SB      1028            // bufB row stride (floats)
#define NT      256             // threads per block
#define NW      8               // waves per block (wave32)

// out[b, g, r, d] = sum_c T[g,r,c,d] * in[b, g, c, d]
//   in  index = g*COL*DIAG + c*DIAG + d
//   out index = g*ROW*DIAG + r*DIAG + d
//   T index   = (s*ROW + r)*COL + c            with s = g*DIAG + d
// WMMA mapping: M = 16 batch rows, N = r (lanes), K = c (chunks of 4).
template <int ROW, int COL, int DIAG, int G>
__device__ __forceinline__ void debut_factor(const float* tws,   // LDS slab
                                             const float* inb, float* outb,
                                             int inStride, int outStride,
                                             int wave, int lane) {
    const int half = lane >> 4;   // A/B K-pair select (ISA 16x4 f32 layout)
    const int lm   = lane & 15;   // M row for A, N column (=r) for B/D
    constexpr int NSUB = G * DIAG;            // 512/256/256/64 -> all %8 == 0
    for (int s = wave; s < NSUB; s += NW) {   // wave-uniform trip count
        const int g = s >> __builtin_ctz(DIAG == 0 ? 1 : DIAG); // s / DIAG
        const int d = s & (DIAG - 1);
        const int inBase  = g * (COL * DIAG) + d;
        const int outBase = g * (ROW * DIAG) + d;
        v8f acc = {0.f, 0.f, 0.f, 0.f, 0.f, 0.f, 0.f, 0.f};
#pragma unroll
        for (int kk = 0; kk < COL / 4; ++kk) {
            const int c0 = kk * 4 + 2 * half; // lo lanes: K=c0,c0+1; hi: +2
            v2f a, b;
            // A: 16x4 f32 slab from LDS (lane = batch row, vgpr pair = K)
            a.x = inb[lm * inStride + inBase + (c0    ) * DIAG];
            a.y = inb[lm * inStride + inBase + (c0 + 1) * DIAG];
            // B: 4x16 twiddle slab from LDS staging buffer
            if (lm < ROW) {
                const float* tp = tws + (s * ROW + lm) * COL + c0;
                b.x = tp[0];
                b.y = tp[1];
            } else {
                b.x = 0.f;
                b.y = 0.f;
            }
            // D = A*B + C ; EXEC all-1s here (uniform control flow)
            acc = __builtin_amdgcn_wmma_f32_16x16x4_f32(
                false, a, false, b, (short)0, acc, false, false);
        }
        // D layout: lane -> N=r, vgpr v -> M=v+8*half (batch row)
        if (lm < ROW) {
#pragma unroll
            for (int v = 0; v < 8; ++v)
                outb[(v + 8 * half) * outStride + outBase + lm * DIAG] = acc[v];
        }
    }
}

// Bulk coalesced global -> LDS copy of one twiddle slab (nfloats % 1024 == 0).
__device__ __forceinline__ void stage_tw(float* dst, const float* __restrict__ src,
                                         int nfloats, int tid) {
    const float4* s4 = (const float4*)src;
    float4*       d4 = (float4*)dst;
    const int nv = nfloats >> 2;
    for (int i = tid; i < nv; i += NT) d4[i] = s4[i];
}

__global__ __launch_bounds__(NT)
void debut_conv1d_kernel(const float* __restrict__ x,
                         const float* __restrict__ tw,
                         const float* __restrict__ bias,
                         float* __restrict__ out) {
    __shared__ float bufA[16 * SA];   // 131.3 KB : im2col + F1/F2 in-place
    __shared__ float bufB[16 * SB];   //  65.8 KB : F3 out, F4 in-place
    __shared__ float tws[16384];      //  64.0 KB : staged twiddle slab
                                      // total ~256.5 KB <= 320 KB/WGP

    const int tile = blockIdx.x;      // 16 consecutive im2col rows
    const int tid  = threadIdx.x;
    const int wave = tid >> 5;
    const int lane = tid & 31;

    // ---- im2col tile load: bufA[m][k*512+c] = x[b, c, l+k], k=0..3 ------
    // One thread owns (m,c): 4 consecutive x floats -> 4 strided LDS slots.
    for (int idx = tid; idx < 16 * 512; idx += NT) {
        const int m = idx >> 9;
        const int c = idx & 511;
        const int n = tile * 16 + m;
        const int b = n / L_OUT;
        const int l = n - b * L_OUT;
        const float* xp  = x + ((size_t)(b * 512 + c)) * 2048 + l;
        float*       dst = &bufA[m * SA + c];
        dst[0]    = xp[0];
        dst[512]  = xp[1];
        dst[1024] = xp[2];
        dst[1536] = xp[3];
    }
    stage_tw(tws, tw, 8192, tid);                       // F1 slab (32 KB)
    __syncthreads();

    debut_factor<4, 4, 1, 512>(tws, bufA, bufA, SA, SA, wave, lane);
    __syncthreads();
    stage_tw(tws, tw + 8192, 16384, tid);               // F2 slab (64 KB)
    __syncthreads();
    debut_factor<8, 8, 4, 64>(tws, bufA, bufA, SA, SA, wave, lane);
    __syncthreads();
    stage_tw(tws, tw + 24576, 8192, tid);               // F3 slab (32 KB)
    __syncthreads();
    debut_factor<4, 8, 32, 8>(tws, bufA, bufB, SA, SB, wave, lane);
    __syncthreads();
    stage_tw(tws, tw + 32768, 8192, tid);               // F4 slab (32 KB)
    __syncthreads();
    debut_factor<8, 16, 64, 1>(tws, bufB, bufB, SB, SB, wave, lane);
    __syncthreads();

    // ---- bias add + store: out[b, oc, l] = bufB[m][oc] + bias[oc] -------
    for (int idx = tid; idx < 16 * 512; idx += NT) {
        const int m  = idx >> 9;
        const int oc = idx & 511;
        const int n  = tile * 16 + m;
        const int b  = n / L_OUT;
        const int l  = n - b * L_OUT;
        out[((size_t)(b * 512 + oc)) * L_OUT + l] = bufB[m * SB + oc] + bias[oc];
    }
}

extern "C" void kernel_launch(void* const* d_in, const int* in_sizes, int n_in,
                              void* d_out, int out_size, void* d_ws, size_t ws_size,
                              hipStream_t stream) {
    (void)in_sizes; (void)n_in; (void)d_ws; (void)ws_size; (void)out_size;
    const float* x    = (const float*)d_in[0];
    const float* tw   = (const float*)d_in[1];
    const float* bias = (const float*)d_in[2];
    float*       out  = (float*)d_out;
    debut_conv1d_kernel<<<N_TILES, NT, 0, stream>>>(x, tw, bias, out);
}